// DistSageConv_76209899700290
// MI455X (gfx1250) — compile-verified
//
#include <hip/hip_runtime.h>

typedef float v2f __attribute__((ext_vector_type(2)));
typedef float v8f __attribute__((ext_vector_type(8)));

#define D 128
#define NTILES (D / 16)   // 8 column tiles of the 128-wide output
#define EPW 128           // edges per wave in the aggregation kernel
#define EPT 32            // edges per thread in the degree kernel

// ---------------------------------------------------------------------------
// Kernel 0: zero the aggregation buffer (d_out) and the degree array (d_ws)
// ---------------------------------------------------------------------------
__global__ void k_zero(float4* __restrict__ out4, int n_out4,
                       float* __restrict__ deg, int n_deg) {
    int i = blockIdx.x * blockDim.x + threadIdx.x;
    int stride = gridDim.x * blockDim.x;
    for (int j = i; j < n_out4; j += stride) out4[j] = make_float4(0.f, 0.f, 0.f, 0.f);
    for (int j = i; j < n_deg; j += stride) deg[j] = 0.f;
}

// ---------------------------------------------------------------------------
// Kernel 1: in-degree per dst node. dst_idx is sorted, so each thread walks a
// contiguous chunk and emits one atomic per run boundary.
// ---------------------------------------------------------------------------
__global__ void k_deg(const int* __restrict__ dst, float* __restrict__ deg, int E) {
    long t = (long)blockIdx.x * blockDim.x + threadIdx.x;
    long start = t * EPT;
    if (start >= E) return;
    long end = start + EPT; if (end > E) end = E;
    int cur = dst[start];
    float cnt = 0.f;
    for (long e = start; e < end; ++e) {
        int d = dst[e];
        if (d != cur) { atomicAdd(&deg[cur], cnt); cur = d; cnt = 0.f; }
        cnt += 1.f;
    }
    atomicAdd(&deg[cur], cnt);
}

// ---------------------------------------------------------------------------
// Kernel 2: edge aggregation. One wave per EPW-edge chunk; lane L owns floats
// [4L, 4L+4) of the 128-dim feature row (coalesced 512B gathers).
// src/dst indices are loaded vectorized (one per lane) and broadcast with
// __shfl, so the gather loads can be issued deep ahead of the accumulate.
// Sorted dst => runs flush once; only runs that can straddle a chunk boundary
// (node == first dst of chunk, or the final run) need f32 atomics — interior
// runs are provably private to this chunk and use plain float4 stores onto
// the zeroed buffer.
// ---------------------------------------------------------------------------
#define FLUSH_RUN(node, use_atomic) do {                                      \
    float* p = agg + (size_t)(node) * D + lane * 4;                           \
    if (use_atomic) {                                                         \
        atomicAdd(p + 0, acc.x); atomicAdd(p + 1, acc.y);                     \
        atomicAdd(p + 2, acc.z); atomicAdd(p + 3, acc.w);                     \
    } else {                                                                  \
        *(float4*)p = acc;                                                    \
    }                                                                         \
} while (0)

#define EDGE_STEP(i) do {                                                     \
    int d = __shfl(myd, (i));                                                 \
    int s = __shfl(mys, (i));                                                 \
    const float4* rowp = (const float4*)(x + (size_t)s * D);                  \
    float4 v = rowp[lane];                                                    \
    if (d != cur) {                                                           \
        FLUSH_RUN(cur, cur == firstDst);                                      \
        acc = make_float4(0.f, 0.f, 0.f, 0.f);                                \
        cur = d;                                                              \
    }                                                                         \
    acc.x += v.x; acc.y += v.y; acc.z += v.z; acc.w += v.w;                   \
} while (0)

__global__ void k_edge(const float* __restrict__ x, const int* __restrict__ src,
                       const int* __restrict__ dst, float* __restrict__ agg, int E) {
    int wave = (int)((blockIdx.x * blockDim.x + threadIdx.x) >> 5);
    int lane = threadIdx.x & 31;
    long start = (long)wave * EPW;
    if (start >= E) return;
    long end = start + EPW; if (end > E) end = E;

    int firstDst = dst[start];
    int cur = firstDst;
    float4 acc = make_float4(0.f, 0.f, 0.f, 0.f);

    for (long base = start; base < end; base += 32) {
        int nv = (int)((end - base) < 32 ? (end - base) : 32);
        long le = base + (lane < nv ? lane : (nv - 1));
        int myd = dst[le];                 // vectorized index loads
        int mys = src[le];
        if (nv == 32) {
#pragma unroll
            for (int i = 0; i < 32; ++i) EDGE_STEP(i);
        } else {
            for (int i = 0; i < nv; ++i) EDGE_STEP(i);
        }
    }
    FLUSH_RUN(cur, true);                  // last run may straddle the boundary
}

// ---------------------------------------------------------------------------
// Kernel 3: out = (agg/deg) @ W1^T + x @ W2^T, fp32 WMMA (16x16x4_f32).
// One wave per 16-row m-tile; 8 n-tile accumulators held live (64 VGPRs).
// All 16 B fragments of a K-step are loaded into registers first, a
// sched_barrier stops the scheduler from sinking them, then 16 WMMAs drain
// with counting loadcnt waits instead of per-WMMA full stalls.
// Epilogue: unguarded clause-able stores when the whole tile is in range.
// ---------------------------------------------------------------------------
__global__ __launch_bounds__(256)
void k_gemm(const float* __restrict__ x, const float* __restrict__ W1,
            const float* __restrict__ W2, const float* __restrict__ deg,
            float* __restrict__ out, int N) {
    int wave = (int)((blockIdx.x * blockDim.x + threadIdx.x) >> 5);
    int lane = threadIdx.x & 31;
    int mtiles = (N + 15) / 16;
    if (wave >= mtiles) return;                 // wave-uniform exit

    int row0 = wave * 16;
    int r    = lane & 15;
    int half = lane >> 4;
    int rowA = row0 + r; if (rowA > N - 1) rowA = N - 1;

    float dg  = deg[rowA];
    float scl = 1.0f / fmaxf(dg, 1.0f);

    const float* xrow = x   + (size_t)rowA * D + half * 2;
    const float* grow = out + (size_t)rowA * D + half * 2;   // agg (in d_out)

    v8f acc[NTILES];
#pragma unroll
    for (int t = 0; t < NTILES; ++t) { v8f z = {}; acc[t] = z; }

    for (int k0 = 0; k0 < D; k0 += 4) {
        v2f a_x  = *(const v2f*)(xrow + k0);
        v2f a_gr = *(const v2f*)(grow + k0);
        v2f b1[NTILES], b2[NTILES];
#pragma unroll
        for (int t = 0; t < NTILES; ++t) {
            b1[t] = *(const v2f*)(W1 + (size_t)(t * 16 + r) * D + half * 2 + k0);
            b2[t] = *(const v2f*)(W2 + (size_t)(t * 16 + r) * D + half * 2 + k0);
        }
        v2f a_g; a_g.x = a_gr.x * scl; a_g.y = a_gr.y * scl;
        __builtin_amdgcn_sched_barrier(0);      // keep the load batch ahead of WMMAs
#pragma unroll
        for (int t = 0; t < NTILES; ++t)
            acc[t] = __builtin_amdgcn_wmma_f32_16x16x4_f32(
                false, a_g, false, b1[t], (short)0, acc[t], false, false);
#pragma unroll
        for (int t = 0; t < NTILES; ++t)
            acc[t] = __builtin_amdgcn_wmma_f32_16x16x4_f32(
                false, a_x, false, b2[t], (short)0, acc[t], false, false);
    }

    // D layout: VGPR j -> row = row0 + (lane>>4)*8 + j, col = t*16 + (lane&15)
    int rowD = row0 + half * 8;
    if (row0 + 16 <= N) {
        // whole tile in range: plain stores, no per-element exec predication
#pragma unroll
        for (int j = 0; j < 8; ++j) {
#pragma unroll
            for (int t = 0; t < NTILES; ++t)
                out[(size_t)(rowD + j) * D + t * 16 + r] = acc[t][j];
        }
    } else {
#pragma unroll
        for (int j = 0; j < 8; ++j) {
            int rr = rowD + j;
            if (rr < N) {
#pragma unroll
                for (int t = 0; t < NTILES; ++t)
                    out[(size_t)rr * D + t * 16 + r] = acc[t][j];
            }
        }
    }
}

// ---------------------------------------------------------------------------
extern "C" void kernel_launch(void* const* d_in, const int* in_sizes, int n_in,
                              void* d_out, int out_size, void* d_ws, size_t ws_size,
                              hipStream_t stream) {
    const float* x   = (const float*)d_in[0];
    const float* W1  = (const float*)d_in[1];
    const float* W2  = (const float*)d_in[2];
    const int*   src = (const int*)d_in[3];
    const int*   dst = (const int*)d_in[4];

    int N = in_sizes[0] / D;      // 100000
    int E = in_sizes[3];          // 1600000
    float* out = (float*)d_out;
    float* deg = (float*)d_ws;    // N floats (400 KB)

    // 0) zero agg (in d_out) and deg
    int n_out4 = N * D / 4;
    k_zero<<<2048, 256, 0, stream>>>((float4*)out, n_out4, deg, N);

    // 1) degree histogram over sorted dst
    {
        long threads = ((long)E + EPT - 1) / EPT;
        int blocks = (int)((threads + 255) / 256);
        k_deg<<<blocks, 256, 0, stream>>>(dst, deg, E);
    }

    // 2) edge gather + segment-sum into d_out
    {
        long waves = ((long)E + EPW - 1) / EPW;
        long threads = waves * 32;
        int blocks = (int)((threads + 255) / 256);
        k_edge<<<blocks, 256, 0, stream>>>(x, src, dst, out, E);
    }

    // 3) fused dual GEMM with fp32 WMMA, in-place over d_out
    {
        int mtiles = (N + 15) / 16;
        int blocks = (mtiles + 7) / 8;          // 8 waves per 256-thread block
        k_gemm<<<blocks, 256, 0, stream>>>(x, W1, W2, deg, out, N);
    }
}